// AttnDecoderRNN_73194832658729
// MI455X (gfx1250) — compile-verified
//
#include <hip/hip_runtime.h>
#include <math.h>

typedef __attribute__((ext_vector_type(2))) float v2f;
typedef __attribute__((ext_vector_type(8))) float v8f;

#define B_ 32
#define S_ 41
#define H_ 1024
#define V_ 16384
#define T_ 41

// ---------------------------------------------------------------------------
// Skinny GEMM on the WMMA pipe, K fixed at 1024 (all call sites):
//   C[M,N] = A[M,K] * W[N,K]^T (+ bias[N]) (+ C if accumulate)
// - A panel (16*MTILES rows x 256-float K-chunks) is staged into LDS with
//   GLOBAL_LOAD_ASYNC_TO_LDS_B128 (ASYNCcnt) and read back as ds_load_b64,
//   so the 4 waves of a block share one copy of A.
// - W fragment loads are software-pipelined one k-step ahead so the WMMAs
//   overlap the global loads instead of s_wait_loadcnt 0x0 on each one.
// - Wave tile (16*MTILES) x 64: each W fragment feeds MTILES WMMAs; with
//   M=32 (MTILES=2) W is read exactly once per GEMM = the traffic floor.
// Fragment layouts per CDNA5 ISA 7.12.2 (32-bit A 16x4 / B 4x16 / D 16x16).
// ---------------------------------------------------------------------------
template <int MTILES>
__global__ __launch_bounds__(128) void wmma_gemm_f32(
    const float* __restrict__ A, int lda,
    const float* __restrict__ W, int ldw,
    const float* __restrict__ bias,
    float* __restrict__ C, long long ldc,
    int M, int N, int accumulate)
{
    constexpr int K      = 1024;
    constexpr int ROWS   = 16 * MTILES;
    constexpr int KCHUNK = 256;                     // floats per staged chunk
    __shared__ float sA[ROWS * KCHUNK];             // 16KB (MTILES=1) / 32KB (MTILES=2)

    const int tid   = threadIdx.x;
    const int lane  = tid & 31;
    const int wave  = tid >> 5;                     // 4 waves / block
    const int nbase = (blockIdx.x * 4 + wave) * 64;
    if (nbase >= N) return;                         // wave-uniform (never hit: N % 256 == 0)
    const int m0  = blockIdx.y * ROWS;
    const int row = lane & 15;                      // A: M index / B: N index
    const int kh  = lane >> 4;                      // K pair {0,1} vs {2,3}
    const int col = row;                            // D: N index

    const float* Wrow[4];
#pragma unroll
    for (int j = 0; j < 4; ++j)
        Wrow[j] = W + (long long)(nbase + 16 * j + row) * ldw;

    v8f acc[MTILES][4];
#pragma unroll
    for (int mt = 0; mt < MTILES; ++mt)
#pragma unroll
        for (int j = 0; j < 4; ++j) {
            if (accumulate) {
#pragma unroll
                for (int r = 0; r < 8; ++r) {
                    int m = m0 + mt * 16 + r + 8 * kh;
                    if (m >= M) m = M - 1;
                    acc[mt][j][r] = C[(long long)m * ldc + nbase + 16 * j + col];
                }
            } else {
                v8f z = {0.f, 0.f, 0.f, 0.f, 0.f, 0.f, 0.f, 0.f};
                acc[mt][j] = z;
            }
        }

    const float* Apanel = A + (long long)m0 * lda;
    unsigned sAbase = (unsigned)(size_t)(void*)sA;  // flat->LDS: low 32 bits are the LDS offset

    // W pipeline prologue (k = 0)
    v2f bcur[4];
#pragma unroll
    for (int j = 0; j < 4; ++j) {
        bcur[j].x = Wrow[j][2 * kh];
        bcur[j].y = Wrow[j][2 * kh + 1];
    }

    for (int kc = 0; kc < K; kc += KCHUNK) {
        // -------- async-stage the A chunk (rows m0..m0+ROWS-1, cols kc..kc+KCHUNK-1)
        // ROWS*KCHUNK/4 16-byte groups; trip count is uniform across the block.
        for (int g = tid; g < ROWS * KCHUNK / 4; g += 128) {
            int r  = g / (KCHUNK / 4);
            int c4 = g - r * (KCHUNK / 4);
            const float* gp = Apanel + (long long)r * lda + kc + c4 * 4;
            unsigned laddr  = sAbase + (unsigned)(r * KCHUNK + c4 * 4) * 4u;
            asm volatile("global_load_async_to_lds_b128 %0, %1, off"
                         :: "v"(laddr), "v"(gp) : "memory");
        }
        asm volatile("s_wait_asynccnt 0x0" ::: "memory");
        __syncthreads();

        // -------- K loop over the chunk, W loads pipelined one step ahead
        for (int k = kc; k < kc + KCHUNK; k += 4) {
            int kn = (k + 4 < K) ? (k + 4) : k;     // clamp: last step reloads (harmless)
            v2f bnxt[4];
#pragma unroll
            for (int j = 0; j < 4; ++j) {
                bnxt[j].x = Wrow[j][kn + 2 * kh];
                bnxt[j].y = Wrow[j][kn + 2 * kh + 1];
            }
            v2f a[MTILES];
#pragma unroll
            for (int mt = 0; mt < MTILES; ++mt)
                a[mt] = *(const v2f*)&sA[(mt * 16 + row) * KCHUNK + (k - kc) + 2 * kh];
#pragma unroll
            for (int j = 0; j < 4; ++j)
#pragma unroll
                for (int mt = 0; mt < MTILES; ++mt)
                    acc[mt][j] = __builtin_amdgcn_wmma_f32_16x16x4_f32(
                        false, a[mt], false, bcur[j], (short)0, acc[mt][j], false, false);
#pragma unroll
            for (int j = 0; j < 4; ++j) bcur[j] = bnxt[j];
        }
        __syncthreads();                            // chunk fully consumed before restage
    }

#pragma unroll
    for (int mt = 0; mt < MTILES; ++mt)
#pragma unroll
        for (int j = 0; j < 4; ++j) {
            int   n  = nbase + 16 * j + col;
            float bv = bias ? bias[n] : 0.f;
#pragma unroll
            for (int r = 0; r < 8; ++r) {
                int m = m0 + mt * 16 + r + 8 * kh;
                if (m < M) C[(long long)m * ldc + n] = acc[mt][j][r] + bv;
            }
        }
}

// h0 = encoder_hidden[0], c0 = 0  (must re-init every call: ws is poisoned)
__global__ void init_hc_kernel(const float* __restrict__ eh,
                               float* __restrict__ h, float* __restrict__ c, int n)
{
    int i = blockIdx.x * blockDim.x + threadIdx.x;
    if (i < n) { h[i] = eh[i]; c[i] = 0.f; }
}

// x[b,:] = emb[tok]; tok = (t==0 ? SOS=0 : target[b, t-1])
__global__ void gather_emb_kernel(const float* __restrict__ emb,
                                  const int* __restrict__ target,
                                  float* __restrict__ x, int t)
{
    int b   = blockIdx.x;
    int tok = (t == 0) ? 0 : target[b * T_ + (t - 1)];
    const float* src = emb + (long long)tok * H_;
    float*       dst = x + (long long)b * H_;
    for (int h = threadIdx.x; h < H_; h += blockDim.x) dst[h] = src[h];
}

// Per batch row: scores = tanh(q + keys_proj) . Vw + bV ; softmax ; ctx = w . enc
__global__ __launch_bounds__(256) void attention_kernel(
    const float* __restrict__ q, const float* __restrict__ keys,
    const float* __restrict__ enc, const float* __restrict__ Vw,
    const float* __restrict__ bV,
    float* __restrict__ ctx, float* __restrict__ attn_out, int t)
{
    __shared__ float sw[S_];
    int b = blockIdx.x;
    int tid = threadIdx.x, lane = tid & 31, wave = tid >> 5;   // 8 waves
    const float* qb = q + (long long)b * H_;

    for (int s = wave; s < S_; s += 8) {
        const float* kp = keys + ((long long)b * S_ + s) * H_;
        float acc = 0.f;
        for (int h = lane; h < H_; h += 32)
            acc += tanhf(qb[h] + kp[h]) * Vw[h];
        for (int off = 16; off > 0; off >>= 1)
            acc += __shfl_xor(acc, off, 32);
        if (lane == 0) sw[s] = acc + bV[0];
    }
    __syncthreads();
    if (tid == 0) {                                    // 41-wide softmax, serial is cheap
        float mx = sw[0];
        for (int s = 1; s < S_; ++s) mx = fmaxf(mx, sw[s]);
        float sum = 0.f;
        for (int s = 0; s < S_; ++s) { float e = expf(sw[s] - mx); sw[s] = e; sum += e; }
        float inv = 1.f / sum;
        for (int s = 0; s < S_; ++s) sw[s] *= inv;
    }
    __syncthreads();
    if (tid < S_) attn_out[((long long)b * T_ + t) * S_ + tid] = sw[tid];
    for (int h = tid; h < H_; h += 256) {
        float acc = 0.f;
        const float* eb = enc + (long long)b * S_ * H_ + h;
        for (int s = 0; s < S_; ++s) acc += sw[s] * eb[(long long)s * H_];
        ctx[(long long)b * H_ + h] = acc;
    }
}

// LSTM pointwise: gates = [i|f|g|o] pre-activations (PyTorch order)
__global__ void lstm_kernel(const float* __restrict__ gates,
                            float* __restrict__ h, float* __restrict__ c)
{
    int i = blockIdx.x * blockDim.x + threadIdx.x;
    if (i >= B_ * H_) return;
    int b = i / H_, hh = i - b * H_;
    const float* g = gates + (long long)b * 4 * H_;
    float ig = 1.f / (1.f + expf(-g[hh]));
    float fg = 1.f / (1.f + expf(-g[H_ + hh]));
    float gg = tanhf(g[2 * H_ + hh]);
    float og = 1.f / (1.f + expf(-g[3 * H_ + hh]));
    float cn = fg * c[i] + ig * gg;
    c[i] = cn;
    h[i] = og * tanhf(cn);
}

// In-place log_softmax over a V=16384 row per block
__global__ __launch_bounds__(256) void logsoftmax_kernel(float* __restrict__ p0)
{
    __shared__ float red[8];
    float* p  = p0 + (long long)blockIdx.x * V_;
    int tid = threadIdx.x, lane = tid & 31, wave = tid >> 5;

    float mx = -3.402823466e38f;
    for (int i = tid; i < V_; i += 256) mx = fmaxf(mx, p[i]);
    for (int off = 16; off > 0; off >>= 1) mx = fmaxf(mx, __shfl_xor(mx, off, 32));
    if (lane == 0) red[wave] = mx;
    __syncthreads();
    if (tid == 0) { float m = red[0]; for (int i = 1; i < 8; ++i) m = fmaxf(m, red[i]); red[0] = m; }
    __syncthreads();
    mx = red[0];
    __syncthreads();

    float sum = 0.f;
    for (int i = tid; i < V_; i += 256) sum += expf(p[i] - mx);
    for (int off = 16; off > 0; off >>= 1) sum += __shfl_xor(sum, off, 32);
    if (lane == 0) red[wave] = sum;
    __syncthreads();
    if (tid == 0) { float s = 0.f; for (int i = 0; i < 8; ++i) s += red[i]; red[0] = s; }
    __syncthreads();
    float lse = mx + logf(red[0]);
    for (int i = tid; i < V_; i += 256) p[i] = p[i] - lse;
}

__global__ void copy_hc_kernel(const float* __restrict__ h, const float* __restrict__ c,
                               float* __restrict__ oh, float* __restrict__ oc)
{
    int i = blockIdx.x * blockDim.x + threadIdx.x;
    if (i < B_ * H_) { oh[i] = h[i]; oc[i] = c[i]; }
}

extern "C" void kernel_launch(void* const* d_in, const int* in_sizes, int n_in,
                              void* d_out, int out_size, void* d_ws, size_t ws_size,
                              hipStream_t stream)
{
    const float* enc    = (const float*)d_in[0];   // [B,S,H]
    const float* ehid   = (const float*)d_in[1];   // [1,B,H]
    const int*   target = (const int*)d_in[2];     // [B,T]
    const float* emb    = (const float*)d_in[3];   // [V,H]
    const float* Wq     = (const float*)d_in[4];
    const float* bq     = (const float*)d_in[5];
    const float* Wk     = (const float*)d_in[6];
    const float* bk     = (const float*)d_in[7];
    const float* Vw     = (const float*)d_in[8];
    const float* bV     = (const float*)d_in[9];
    const float* Wih    = (const float*)d_in[10];  // [4H,2H]
    const float* Whh    = (const float*)d_in[11];  // [4H,H]
    const float* bih    = (const float*)d_in[12];
    const float* bhh    = (const float*)d_in[13];
    const float* Wout   = (const float*)d_in[14];  // [V,H]
    const float* bout   = (const float*)d_in[15];

    float* ws    = (float*)d_ws;
    float* keys  = ws;  ws += (long long)B_ * S_ * H_;  // keys_proj [B,S,H]
    float* h     = ws;  ws += B_ * H_;
    float* c     = ws;  ws += B_ * H_;
    float* q     = ws;  ws += B_ * H_;
    float* x     = ws;  ws += B_ * H_;
    float* ctx   = ws;  ws += B_ * H_;
    float* gates = ws;  ws += B_ * 4 * H_;

    float* out_logits = (float*)d_out;                       // [B,T,V]
    float* out_h      = out_logits + (long long)B_ * T_ * V_;
    float* out_c      = out_h + B_ * H_;
    float* out_attn   = out_c + B_ * H_;                     // [B,T,S]

    init_hc_kernel<<<(B_ * H_ + 255) / 256, 256, 0, stream>>>(ehid, h, c, B_ * H_);

    // keys_proj = enc @ Wk^T + bk   (M=1312, N=1024, K=1024)
    wmma_gemm_f32<1><<<dim3(H_ / 256, (B_ * S_) / 16), 128, 0, stream>>>(
        enc, H_, Wk, H_, bk, keys, H_, B_ * S_, H_, 0);

    for (int t = 0; t < T_; ++t) {
        gather_emb_kernel<<<B_, 256, 0, stream>>>(emb, target, x, t);

        // q = h @ Wq^T + bq
        wmma_gemm_f32<2><<<dim3(H_ / 256, 1), 128, 0, stream>>>(
            h, H_, Wq, H_, bq, q, H_, B_, H_, 0);

        attention_kernel<<<B_, 256, 0, stream>>>(q, keys, enc, Vw, bV, ctx, out_attn, t);

        // gates = x @ Wih[:,:H]^T + bih
        wmma_gemm_f32<2><<<dim3(4 * H_ / 256, 1), 128, 0, stream>>>(
            x, H_, Wih, 2 * H_, bih, gates, 4 * H_, B_, 4 * H_, 0);
        // gates += ctx @ Wih[:,H:]^T + bhh
        wmma_gemm_f32<2><<<dim3(4 * H_ / 256, 1), 128, 0, stream>>>(
            ctx, H_, Wih + H_, 2 * H_, bhh, gates, 4 * H_, B_, 4 * H_, 1);
        // gates += h @ Whh^T
        wmma_gemm_f32<2><<<dim3(4 * H_ / 256, 1), 128, 0, stream>>>(
            h, H_, Whh, H_, nullptr, gates, 4 * H_, B_, 4 * H_, 1);

        lstm_kernel<<<(B_ * H_ + 255) / 256, 256, 0, stream>>>(gates, h, c);

        // logits_t = h_new @ Wout^T + bout, written directly into d_out[B,T,V]
        wmma_gemm_f32<2><<<dim3(V_ / 256, 1), 128, 0, stream>>>(
            h, H_, Wout, H_, bout, out_logits + (long long)t * V_,
            (long long)T_ * V_, B_, V_, 0);
    }

    logsoftmax_kernel<<<B_ * T_, 256, 0, stream>>>(out_logits);
    copy_hc_kernel<<<(B_ * H_ + 255) / 256, 256, 0, stream>>>(h, c, out_h, out_c);
}